// SurfaceDiceLoss_13546326851822
// MI455X (gfx1250) — compile-verified
//
#include <hip/hip_runtime.h>
#include <math.h>

// ---------------- problem geometry (fixed by the reference) ----------------
constexpr int Bn = 2, Zn = 9, Yn = 384, Xn = 384;
constexpr int ZP = 8;              // z-pairs = Z-1
constexpr int OH = 385, OW = 385;  // unfold output = (Y+1) x (X+1)
constexpr int SITES = OH * OW;                 // 148225
constexpr int TOT   = Bn * ZP * SITES;         // 2,371,600 sites
constexpr float SMOOTH = 0.001f;

constexpr int NBLK = 1024;   // main-kernel grid (also rows of the partial table)
constexpr int NTHR = 256;    // 8 waves/block on wave32

typedef __attribute__((ext_vector_type(2))) float v2f;
typedef __attribute__((ext_vector_type(8))) float v8f;

__device__ __forceinline__ float sigmoid_f(float v) {
    return 1.0f / (1.0f + __expf(-v));
}

// ---------------------------------------------------------------------------
// Kernel 1: fused sigmoid + 2x2x2 unfold + 8-step simplex solver + reduction
// to per-block partial sums {num_b0, den_b0, num_b1, den_b1} in ws.
// ---------------------------------------------------------------------------
__global__ __launch_bounds__(NTHR)
void sd_main(const float* __restrict__ pred, const int* __restrict__ labels,
             const float* __restrict__ area, float* __restrict__ ws) {
    __shared__ float s_area[256];
    __shared__ float s_red[4 * NTHR];

    s_area[threadIdx.x] = area[threadIdx.x];   // 256-entry LUT -> LDS
    __syncthreads();

    float n0 = 0.f, d0 = 0.f, n1 = 0.f, d1 = 0.f;

    for (int t = blockIdx.x * NTHR + threadIdx.x; t < TOT; t += NBLK * NTHR) {
        int xp = t % OW;
        int r  = t / OW;
        int yp = r % OH;  r /= OH;
        int zp = r % ZP;
        int b  = r / ZP;

        float xv[8];
        int lbyte = 0;
        #pragma unroll
        for (int c = 0; c < 2; ++c) {
            const int z = zp + c;
            const float* pb = pred   + ((b * Zn + z) * Yn) * Xn;
            const int*   lb = labels + ((b * Zn + z) * Yn) * Xn;
            #pragma unroll
            for (int kh = 0; kh < 2; ++kh) {
                const int y = yp + kh - 1;
                #pragma unroll
                for (int kw = 0; kw < 2; ++kw) {
                    const int x = xp + kw - 1;
                    const int i = c * 4 + kh * 2 + kw;   // channel-major, torch unfold order
                    float pv = 0.f;
                    int   lv = 0;
                    if (y >= 0 && y < Yn && x >= 0 && x < Xn) {
                        const int o = y * Xn + x;
                        pv = sigmoid_f(pb[o]);
                        lv = lb[o];
                    }
                    xv[i] = pv;                 // zero padding (not sigmoid(0))
                    lbyte |= (lv & 1) << i;
                }
            }
        }

        const float la = s_area[lbyte];

        // greedy decomposition: N_SOLVER_STEPS = 8
        float pa = 0.f;
        #pragma unroll 1
        for (int s = 0; s < 8; ++s) {
            int byte = 0;
            float w = 3.402823466e38f;
            #pragma unroll
            for (int i = 0; i < 8; ++i) {
                if (xv[i] > 0.f) { byte |= 1 << i; w = fminf(w, xv[i]); }
            }
            if (byte == 0) break;               // remaining steps contribute 0
            pa += w * s_area[byte];
            #pragma unroll
            for (int i = 0; i < 8; ++i) {
                if ((byte >> i) & 1) xv[i] -= w;  // min entry -> exact 0 (matches jnp)
            }
        }

        const float s2   = la + pa;
        const float both = (pa > 0.f && la > 0.f) ? 1.f : 0.f;
        if (b == 0) { n0 += both * s2; d0 += s2; }
        else        { n1 += both * s2; d1 += s2; }
    }

    // deterministic block tree-reduction
    s_red[0 * NTHR + threadIdx.x] = n0;
    s_red[1 * NTHR + threadIdx.x] = d0;
    s_red[2 * NTHR + threadIdx.x] = n1;
    s_red[3 * NTHR + threadIdx.x] = d1;
    __syncthreads();
    for (int off = NTHR / 2; off > 0; off >>= 1) {
        if (threadIdx.x < off) {
            #pragma unroll
            for (int k = 0; k < 4; ++k)
                s_red[k * NTHR + threadIdx.x] += s_red[k * NTHR + threadIdx.x + off];
        }
        __syncthreads();
    }
    if (threadIdx.x == 0) {
        #pragma unroll
        for (int k = 0; k < 4; ++k)
            ws[blockIdx.x * 4 + k] = s_red[k * NTHR];
    }
}

// ---------------------------------------------------------------------------
// Kernel 2 (one wave32): WMMA-based column-sum of the NBLK x 4 partial table.
// A = ones(16x4), B carries 64 partials of one accumulator class per issue:
//   D[m][n] = sum_k B[k][n] + C[m][n]  -> after all chunks, sum of D's first
// register over all 32 lanes equals 2x the grand total, independent of the
// exact f32 4x16 B lane layout (A is all-ones, each WMMA holds one class).
// EXEC is all-ones here (full wave, uniform flow) as WMMA requires.
// ---------------------------------------------------------------------------
__global__ __launch_bounds__(32)
void sd_final(const float* __restrict__ ws, float* __restrict__ out) {
    const int lane = threadIdx.x;              // 0..31
    v2f ones;
    ones.x = 1.0f; ones.y = 1.0f;

    float tot[4];
    #pragma unroll
    for (int cls = 0; cls < 4; ++cls) {
        v8f acc = {};
        #pragma unroll 1
        for (int ch = 0; ch < NBLK / 64; ++ch) {       // 16 chunks of 64 rows
            v2f bm;
            bm.x = ws[(ch * 64 +      lane) * 4 + cls];
            bm.y = ws[(ch * 64 + 32 + lane) * 4 + cls];
            acc = __builtin_amdgcn_wmma_f32_16x16x4_f32(
                /*neg_a=*/false, ones, /*neg_b=*/false, bm,
                /*c_mod=*/(short)0, acc, /*reuse_a=*/false, /*reuse_b=*/false);
        }
        float v = acc[0];                  // row {0,8} col-sums across lanes
        #pragma unroll
        for (int m = 16; m > 0; m >>= 1)   // wave32 butterfly reduce
            v += __shfl_xor(v, m, 32);
        tot[cls] = v * 0.5f;               // each column-sum counted twice
    }

    if (lane == 0) {
        const float dice0 = 1.f - (tot[0] + SMOOTH) / (tot[1] + SMOOTH);
        const float dice1 = 1.f - (tot[2] + SMOOTH) / (tot[3] + SMOOTH);
        out[0] = 0.5f * (dice0 + dice1);
    }
}

// ---------------------------------------------------------------------------
extern "C" void kernel_launch(void* const* d_in, const int* in_sizes, int n_in,
                              void* d_out, int out_size, void* d_ws, size_t ws_size,
                              hipStream_t stream) {
    const float* pred   = (const float*)d_in[0];  // [2,9,384,384] f32
    const int*   labels = (const int*)  d_in[1];  // [2,9,384,384] i32
    const float* area   = (const float*)d_in[2];  // [256] f32
    float* out = (float*)d_out;                   // scalar f32
    float* ws  = (float*)d_ws;                    // NBLK*4 floats = 16 KB

    sd_main <<<NBLK, NTHR, 0, stream>>>(pred, labels, area, ws);
    sd_final<<<1,    32,   0, stream>>>(ws, out);
}